// KnnModel_31937376813221
// MI455X (gfx1250) — compile-verified
//
#include <hip/hip_runtime.h>
#include <math.h>
#include <float.h>

// ---------------------------------------------------------------------------
// KNN (B=2048, D=128, N=100000, K=5), weights='distance'
//   pass 1: q = (feats-mean)/std, q2 = ||q||^2
//   pass 2: x2 = ||Xn||^2
//   pass 3: fused d2 = q2 + x2 - 2 q.Xn^T via V_WMMA_F32_16X16X4_F32.
//           Workgroup = 8 waves = 128 queries; one 16-point Xn tile staged in
//           LDS via global_load_async_to_lds_b128 (double-buffered, ASYNCcnt),
//           pre-swizzled into WMMA A-fragment order (bank-conflict-free).
//           Streaming per-lane top-5 on d2 (sqrt is monotonic).
//   pass 4: merge NSPLIT partials, w = rsqrt(max(d2,eps)), weighted average
// ---------------------------------------------------------------------------

typedef __attribute__((ext_vector_type(2))) float v2f;
typedef __attribute__((ext_vector_type(8))) float v8f;

#define B_Q    2048
#define D_F    128
#define N_PTS  100000
#define K_NN   5
#define NSPLIT 32
#define QPB    128                       // queries per workgroup (8 waves x 16)
#define QBLOCKS (B_Q / QPB)              // 16
#define NTILES (N_PTS / 16)              // 6250, exact
#define TILES_PER_SPLIT ((NTILES + NSPLIT - 1) / NSPLIT)   // 196
#define EPS_F  1e-12f

// Branchless sorted-ascending top-K insertion (K=5).
__device__ __forceinline__ void topk_insert(float (&d)[K_NN], int (&i)[K_NN],
                                            float nd, int ni) {
    if (nd >= d[K_NN - 1]) return;
#pragma unroll
    for (int j = 0; j < K_NN; ++j) {
        bool lt = nd < d[j];
        float carry_d = lt ? d[j] : nd;
        int   carry_i = lt ? i[j] : ni;
        d[j] = lt ? nd : d[j];
        i[j] = lt ? ni : i[j];
        nd = carry_d;
        ni = carry_i;
    }
}

// --- pass 1: normalize queries + row squared norms (1 wave per row) --------
__global__ __launch_bounds__(256) void knn_normalize(
    const float* __restrict__ feats, const float* __restrict__ means,
    const float* __restrict__ stds, float* __restrict__ q,
    float* __restrict__ q2) {
    const int lane = threadIdx.x & 31;
    const int wave = threadIdx.x >> 5;
    const int row  = blockIdx.x * 8 + wave;          // 256 blocks * 8 = 2048

    float4 m = ((const float4*)means)[lane];         // 32 lanes * 4 = 128
    float4 s = ((const float4*)stds)[lane];
    float4 x = ((const float4*)(feats + (size_t)row * D_F))[lane];
    float4 r;
    r.x = (x.x - m.x) / s.x;
    r.y = (x.y - m.y) / s.y;
    r.z = (x.z - m.z) / s.z;
    r.w = (x.w - m.w) / s.w;
    ((float4*)(q + (size_t)row * D_F))[lane] = r;

    float ss = r.x * r.x + r.y * r.y + r.z * r.z + r.w * r.w;
#pragma unroll
    for (int off = 16; off > 0; off >>= 1) ss += __shfl_xor(ss, off, 32);
    if (lane == 0) q2[row] = ss;
}

// --- pass 2: database row squared norms (1 wave per row) -------------------
__global__ __launch_bounds__(256) void knn_xnorm(
    const float* __restrict__ Xn, float* __restrict__ x2) {
    const int lane = threadIdx.x & 31;
    const int wave = threadIdx.x >> 5;
    const int row  = blockIdx.x * 8 + wave;
    if (row >= N_PTS) return;

    float4 x = ((const float4*)(Xn + (size_t)row * D_F))[lane];
    float ss = x.x * x.x + x.y * x.y + x.z * x.z + x.w * x.w;
#pragma unroll
    for (int off = 16; off > 0; off >>= 1) ss += __shfl_xor(ss, off, 32);
    if (lane == 0) x2[row] = ss;
}

// --- pass 3: fused WMMA GEMM + streaming top-5 -----------------------------
// grid.x = QBLOCKS * NSPLIT = 512; block = 256 (8 waves).
// Wave w owns queries [qb*128 + w*16, +16): B operand resident in 64 VGPRs.
// Per Xn tile (16 points x 128 K, 8 KB):
//   staged to LDS with global_load_async_to_lds_b128, pre-swizzled so that
//   LDS word (s*64 + 4*(lane&15) + 2*(lane>>4)) is lane's A-fragment for
//   K-step s  ->  32 conflict-free ds_load_b64 + 32 wmma per wave per tile.
__global__ __launch_bounds__(256) void knn_gemm_topk(
    const float* __restrict__ q, const float* __restrict__ q2,
    const float* __restrict__ Xn, const float* __restrict__ x2,
    float* __restrict__ cand_d, int* __restrict__ cand_i) {
    __shared__ float smem[2][2048];          // 2 x 8KB tile buffers
    __shared__ float md[256 * K_NN];
    __shared__ int   mi[256 * K_NN];

    const int tid   = threadIdx.x;
    const int lane  = tid & 31;
    const int wave  = tid >> 5;
    const int qb    = blockIdx.x / NSPLIT;
    const int split = blockIdx.x % NSPLIT;

    const int m    = lane & 15;              // query within wave's sub-tile
    const int half = lane >> 4;              // selects K/M offset within frags
    const int qrow = qb * QPB + wave * 16 + m;

    const int tileBeg = split * TILES_PER_SPLIT;
    const int tileCap = tileBeg + TILES_PER_SPLIT;
    const int tileEnd = tileCap < NTILES ? tileCap : NTILES;

    // LDS byte offsets of the two tile buffers (low 32 bits of flat address
    // of a __shared__ object == wave-relative LDS offset on gfx1250).
    uint32_t ldsBase[2];
    ldsBase[0] = (uint32_t)(uintptr_t)&smem[0][0];
    ldsBase[1] = (uint32_t)(uintptr_t)&smem[1][0];

    // Async-stage one 16x128 f32 tile into LDS, swizzled to fragment order:
    // 16B chunk (row, c) of the tile -> LDS offset (c*16 + row)*16.
    // 512 chunks / 256 threads = 2 async b128 copies per thread.
    auto stage = [&](int tile, int buf) {
#pragma unroll
        for (int r = 0; r < 2; ++r) {
            int ch  = tid + r * 256;
            int row = ch & 15;
            int c   = ch >> 4;
            const float* src = Xn + (size_t)(tile * 16 + row) * D_F + c * 4;
            uint32_t dst = ldsBase[buf] + (uint32_t)((c * 16 + row) * 16);
            asm volatile("global_load_async_to_lds_b128 %0, %1, off"
                         :: "v"(dst), "v"(src) : "memory");
        }
    };

    // Load B operand (this wave's query tile) once: bq[s] = q[m][4s+2*half..+1]
    // Issue first tile's async copy before these loads so it overlaps.
    if (tileBeg < tileEnd) stage(tileBeg, 0);

    v2f bq[32];
    {
        const float* qp = q + (size_t)qrow * D_F + 2 * half;
#pragma unroll
        for (int s = 0; s < 32; ++s) bq[s] = *(const v2f*)(qp + 4 * s);
    }
    const float q2m = q2[qrow];

    float td[K_NN];
    int   ti[K_NN];
#pragma unroll
    for (int j = 0; j < K_NN; ++j) { td[j] = FLT_MAX; ti[j] = 0; }

    asm volatile("s_wait_asynccnt 0" ::: "memory");
    __syncthreads();                          // tile 0 resident for all waves

    for (int tile = tileBeg; tile < tileEnd; ++tile) {
        const int cur = (tile - tileBeg) & 1;
        if (tile + 1 < tileEnd) stage(tile + 1, cur ^ 1);   // overlap copy

        const float* abuf = &smem[cur][(size_t)(4 * m + 2 * half)];

        v8f c0 = {}, c1 = {};
#pragma unroll
        for (int s = 0; s < 32; s += 2) {
            v2f a0 = *(const v2f*)(abuf + (s + 0) * 64);
            v2f a1 = *(const v2f*)(abuf + (s + 1) * 64);
            c0 = __builtin_amdgcn_wmma_f32_16x16x4_f32(
                     false, a0, false, bq[s + 0], (short)0, c0, false, false);
            c1 = __builtin_amdgcn_wmma_f32_16x16x4_f32(
                     false, a1, false, bq[s + 1], (short)0, c1, false, false);
        }

        const int nb0 = tile * 16 + 8 * half;
        const float4 xa = *(const float4*)(x2 + nb0);
        const float4 xb = *(const float4*)(x2 + nb0 + 4);
        float xs[8] = {xa.x, xa.y, xa.z, xa.w, xb.x, xb.y, xb.z, xb.w};
#pragma unroll
        for (int v = 0; v < 8; ++v) {
            float d2 = fmaf(-2.0f, c0[v] + c1[v], q2m + xs[v]);
            topk_insert(td, ti, d2, nb0 + v);
        }

        if (tile + 1 < tileEnd)
            asm volatile("s_wait_asynccnt 0" ::: "memory");
        __syncthreads();   // next tile resident; safe to overwrite other buf
    }

    // Merge: lanes l and l+16 of each wave hold the same query.
#pragma unroll
    for (int j = 0; j < K_NN; ++j) {
        md[tid * K_NN + j] = td[j];
        mi[tid * K_NN + j] = ti[j];
    }
    __syncthreads();

    if (tid < 128) {
        const int w = tid >> 4;
        const int mq = tid & 15;
        float fd[K_NN];
        int   fi[K_NN];
#pragma unroll
        for (int j = 0; j < K_NN; ++j) { fd[j] = FLT_MAX; fi[j] = 0; }
        const int sA = (w * 32 + mq) * K_NN;
        const int sB = (w * 32 + 16 + mq) * K_NN;
#pragma unroll
        for (int j = 0; j < K_NN; ++j) topk_insert(fd, fi, md[sA + j], mi[sA + j]);
#pragma unroll
        for (int j = 0; j < K_NN; ++j) topk_insert(fd, fi, md[sB + j], mi[sB + j]);

        const int qg  = qb * QPB + w * 16 + mq;
        const int off = (qg * NSPLIT + split) * K_NN;
#pragma unroll
        for (int j = 0; j < K_NN; ++j) {
            cand_d[off + j] = fd[j];
            cand_i[off + j] = fi[j];
        }
    }
}

// --- pass 4: merge splits + inverse-distance weighting ---------------------
__global__ __launch_bounds__(256) void knn_finalize(
    const float* __restrict__ cand_d, const int* __restrict__ cand_i,
    const float* __restrict__ y, float* __restrict__ out) {
    const int b = blockIdx.x * blockDim.x + threadIdx.x;
    if (b >= B_Q) return;

    float fd[K_NN];
    int   fi[K_NN];
#pragma unroll
    for (int j = 0; j < K_NN; ++j) { fd[j] = FLT_MAX; fi[j] = 0; }
    const int base = b * NSPLIT * K_NN;
    for (int j = 0; j < NSPLIT * K_NN; ++j)
        topk_insert(fd, fi, cand_d[base + j], cand_i[base + j]);

    // w = 1/max(sqrt(max(d2,eps)),eps) == rsqrt(max(d2,eps))
    float sw = 0.0f, swy = 0.0f;
#pragma unroll
    for (int j = 0; j < K_NN; ++j) {
        float w = rsqrtf(fmaxf(fd[j], EPS_F));
        sw  += w;
        swy += w * y[fi[j]];
    }
    out[b] = swy / sw;
}

// ---------------------------------------------------------------------------
extern "C" void kernel_launch(void* const* d_in, const int* in_sizes, int n_in,
                              void* d_out, int out_size, void* d_ws,
                              size_t ws_size, hipStream_t stream) {
    (void)in_sizes; (void)n_in; (void)out_size; (void)ws_size;
    const float* feats = (const float*)d_in[0];   // (B, D)
    const float* means = (const float*)d_in[1];   // (D,)
    const float* stds  = (const float*)d_in[2];   // (D,)
    const float* Xn    = (const float*)d_in[3];   // (N, D)
    const float* y     = (const float*)d_in[4];   // (N,)
    // d_in[5] = k (compile-time K_NN = 5)

    float* ws   = (float*)d_ws;
    float* q    = ws;                              // B*D        = 262144 f
    float* q2   = q  + (size_t)B_Q * D_F;          // B          = 2048 f
    float* x2   = q2 + B_Q;                        // N          = 100000 f
    float* cd   = x2 + N_PTS;                      // B*NSPLIT*K = 327680 f
    int*   ci   = (int*)(cd + (size_t)B_Q * NSPLIT * K_NN);
    float* out  = (float*)d_out;                   // (B, 1)

    knn_normalize<<<B_Q / 8, 256, 0, stream>>>(feats, means, stds, q, q2);
    knn_xnorm<<<(N_PTS + 7) / 8, 256, 0, stream>>>(Xn, x2);
    knn_gemm_topk<<<QBLOCKS * NSPLIT, 256, 0, stream>>>(q, q2, Xn, x2, cd, ci);
    knn_finalize<<<(B_Q + 255) / 256, 256, 0, stream>>>(cd, ci, y, out);
}